// OptimizedPromptGraph_74251394613540
// MI455X (gfx1250) — compile-verified
//
#include <hip/hip_runtime.h>

typedef __attribute__((ext_vector_type(2))) float v2f;
typedef __attribute__((ext_vector_type(8))) float v8f;

#define N_NODES 100000
#define N_EDGES 3200000
#define DIM 128
#define LN_EPS 1e-5f
#define SA_STRIDE 132           // 128 + 4 pad -> WMMA A-frag ds_load_b64 conflict-free
#define ROWBLOCKS (N_NODES/16)  // 6250 exactly

// Force the native no-return f32 atomic (global_atomic_add_f32) instead of a
// possible CAS expansion; scatter-add is the bandwidth/atomic-bound phase.
__device__ __forceinline__ void atom_add_f32(float* p, float v) {
    unsafeAtomicAdd(p, v);
}

// ---------------- degree: integer atomics (global_atomic_add_u32) ----------------
__global__ void k_init_deg(unsigned int* deg) {
    int i = blockIdx.x * blockDim.x + threadIdx.x;
    if (i < N_NODES) deg[i] = 1u;                 // self-loop contributes 1
}

__global__ void k_count_deg(const long long* __restrict__ col, unsigned int* deg) {
    int e = blockIdx.x * blockDim.x + threadIdx.x;
    if (e < N_EDGES) atomicAdd(&deg[(int)col[e]], 1u);
}

__global__ void k_dinv(const unsigned int* __restrict__ deg, float* __restrict__ dinv) {
    int i = blockIdx.x * blockDim.x + threadIdx.x;
    if (i < N_NODES) dinv[i] = rsqrtf((float)deg[i]);   // deg >= 1 always
}

// ---------------- agg init with self-loop term: agg[i] = dinv[i]^2 * x[i] --------
__global__ void k_self(const float* __restrict__ x, const float* __restrict__ dinv,
                       float* __restrict__ agg) {
    int idx = blockIdx.x * blockDim.x + threadIdx.x;   // over float4 elements
    if (idx < N_NODES * (DIM/4)) {
        int node = idx >> 5;                           // 32 float4 per row
        float s = dinv[node]; s *= s;
        float4 xv = ((const float4*)x)[idx];
        float4 o; o.x = s*xv.x; o.y = s*xv.y; o.z = s*xv.z; o.w = s*xv.w;
        ((float4*)agg)[idx] = o;
    }
}

// ---------------- edge scatter: one wave32 per edge, lane owns 4 floats ----------
// x, agg, and edge slices are L2-resident (51 MB each vs 192 MB L2); the wave's
// 32 lanes cover one full 512B row so the per-wave atomic burst is cacheline-
// contiguous and L2 can merge it.
__global__ __launch_bounds__(256) void k_scatter(
        const long long* __restrict__ row,
        const long long* __restrict__ col,
        const float* __restrict__ x,
        const float* __restrict__ dinv,
        float* __restrict__ agg) {
    int e = blockIdx.x * 8 + (threadIdx.x >> 5);       // 8 waves per block
    if (e >= N_EDGES) return;
    int lane = threadIdx.x & 31;
    int r = (int)row[e];
    int c = (int)col[e];
    float norm = dinv[r] * dinv[c];
    int d0 = lane * 4;
    float4 xv = *(const float4*)(x + (long)r * DIM + d0);
    float* dst = agg + (long)c * DIM + d0;
    atom_add_f32(dst + 0, norm * xv.x);
    atom_add_f32(dst + 1, norm * xv.y);
    atom_add_f32(dst + 2, norm * xv.z);
    atom_add_f32(dst + 3, norm * xv.w);
}

// ---------------- fused GEMM (out = agg @ W^T + b) + LayerNorm -------------------
// 256 threads = 8 waves; wave w computes the 16x16 tile covering output cols
// [16w, 16w+16). B (= W^T slice) is register-resident: 32 steps of K=4, one
// float2 fragment per lane per step. A tile (16x128) is staged in LDS.
__global__ __launch_bounds__(256) void k_gemm_ln(
    const float* __restrict__ agg, const float* __restrict__ W,
    const float* __restrict__ bvec, const float* __restrict__ gamma,
    const float* __restrict__ beta, float* __restrict__ out) {

    __shared__ float sA[16 * SA_STRIDE];
    __shared__ float sO[16 * SA_STRIDE];

    const int tid  = threadIdx.x;
    const int lane = tid & 31;
    const int wave = tid >> 5;      // 0..7 -> 16-col slice
    const int hl   = lane >> 4;     // lane half (K offset 0 or 2)
    const int l15  = lane & 15;

    // B[k][n] = W[n0+n][k]; fragment for step s: (K = 4s + 2*hl + {0,1}, N = l15)
    v2f breg[32];
    {
        const float* wp = W + (wave * 16 + l15) * DIM + 2 * hl;
        #pragma unroll
        for (int s = 0; s < 32; ++s)
            breg[s] = *(const v2f*)(wp + 4 * s);
    }
    const float bias = bvec[wave * 16 + l15];

    for (int blk = blockIdx.x; blk < ROWBLOCKS; blk += gridDim.x) {
        // stage agg tile (16 rows x 128) into padded LDS
        const float* src = agg + (long)blk * 16 * DIM;
        #pragma unroll
        for (int t = tid; t < 16 * 32; t += 256) {
            int rr = t >> 5, c4 = (t & 31) * 4;
            *(float4*)(&sA[rr * SA_STRIDE + c4]) = *(const float4*)(src + rr * DIM + c4);
        }
        __syncthreads();

        v8f c = {0.f, 0.f, 0.f, 0.f, 0.f, 0.f, 0.f, 0.f};
        #pragma unroll
        for (int s = 0; s < 32; ++s) {
            // A fragment: (M = l15, K = 4s + 2*hl + {0,1}) -> ds_load_b64
            v2f a = *(const v2f*)(&sA[l15 * SA_STRIDE + 4 * s + 2 * hl]);
            c = __builtin_amdgcn_wmma_f32_16x16x4_f32(
                    false, a, false, breg[s], (short)0, c, false, false);
        }

        // D layout: VGPR v -> (M = v + 8*hl, N = l15); add bias, park in LDS
        #pragma unroll
        for (int v = 0; v < 8; ++v)
            sO[(v + 8 * hl) * SA_STRIDE + wave * 16 + l15] = c[v] + bias;
        __syncthreads();

        // LayerNorm: 16 threads per row, 8 elems each; reduce in wave32 halves
        {
            int r  = tid >> 4;
            int cg = (tid & 15) * 8;
            const float* rp = &sO[r * SA_STRIDE + cg];
            float4 u0 = *(const float4*)(rp);
            float4 u1 = *(const float4*)(rp + 4);
            float sum = u0.x + u0.y + u0.z + u0.w + u1.x + u1.y + u1.z + u1.w;
            float sq  = u0.x*u0.x + u0.y*u0.y + u0.z*u0.z + u0.w*u0.w
                      + u1.x*u1.x + u1.y*u1.y + u1.z*u1.z + u1.w*u1.w;
            #pragma unroll
            for (int m = 1; m <= 8; m <<= 1) {
                sum += __shfl_xor(sum, m, 32);
                sq  += __shfl_xor(sq,  m, 32);
            }
            float mean = sum * (1.0f / DIM);
            float var  = sq  * (1.0f / DIM) - mean * mean;
            float rstd = rsqrtf(var + LN_EPS);
            float4 g0  = *(const float4*)(gamma + cg);
            float4 g1  = *(const float4*)(gamma + cg + 4);
            float4 be0 = *(const float4*)(beta + cg);
            float4 be1 = *(const float4*)(beta + cg + 4);
            float4 o0, o1;
            o0.x = (u0.x - mean) * rstd * g0.x + be0.x;
            o0.y = (u0.y - mean) * rstd * g0.y + be0.y;
            o0.z = (u0.z - mean) * rstd * g0.z + be0.z;
            o0.w = (u0.w - mean) * rstd * g0.w + be0.w;
            o1.x = (u1.x - mean) * rstd * g1.x + be1.x;
            o1.y = (u1.y - mean) * rstd * g1.y + be1.y;
            o1.z = (u1.z - mean) * rstd * g1.z + be1.z;
            o1.w = (u1.w - mean) * rstd * g1.w + be1.w;
            float* op = out + ((long)blk * 16 + r) * DIM + cg;
            *(float4*)(op)     = o0;
            *(float4*)(op + 4) = o1;
        }
        __syncthreads();
    }
}

extern "C" void kernel_launch(void* const* d_in, const int* in_sizes, int n_in,
                              void* d_out, int out_size, void* d_ws, size_t ws_size,
                              hipStream_t stream) {
    const float*     x     = (const float*)d_in[0];
    const long long* eidx  = (const long long*)d_in[1];   // int64 [2, E]
    const float*     W     = (const float*)d_in[2];
    const float*     bvec  = (const float*)d_in[3];
    const float*     gamma = (const float*)d_in[4];
    const float*     beta  = (const float*)d_in[5];
    float*           out   = (float*)d_out;

    const long long* erow = eidx;            // source j
    const long long* ecol = eidx + N_EDGES;  // target i

    // workspace layout: agg[N*D] f32 | deg[N] u32 | dinv[N] f32
    char* ws = (char*)d_ws;
    float*        agg  = (float*)ws;                      ws += (size_t)N_NODES * DIM * 4;
    unsigned int* deg  = (unsigned int*)ws;               ws += (size_t)N_NODES * 4;
    float*        dinv = (float*)ws;

    k_init_deg<<<(N_NODES + 255) / 256, 256, 0, stream>>>(deg);
    k_count_deg<<<(N_EDGES + 255) / 256, 256, 0, stream>>>(ecol, deg);
    k_dinv<<<(N_NODES + 255) / 256, 256, 0, stream>>>(deg, dinv);
    k_self<<<(N_NODES * (DIM/4) + 255) / 256, 256, 0, stream>>>(x, dinv, agg);
    k_scatter<<<N_EDGES / 8, 256, 0, stream>>>(erow, ecol, x, dinv, agg);
    k_gemm_ln<<<ROWBLOCKS / 2, 256, 0, stream>>>(agg, W, bvec, gamma, beta, out);
}